// GPUNUFFTAdj_87565793230871
// MI455X (gfx1250) — compile-verified
//
#include <hip/hip_runtime.h>
#include <math.h>

typedef float v2f __attribute__((ext_vector_type(2)));
typedef float v8f __attribute__((ext_vector_type(8)));
typedef unsigned int u32x4 __attribute__((ext_vector_type(4)));
typedef int i32x4 __attribute__((ext_vector_type(4)));
typedef int i32x8 __attribute__((ext_vector_type(8)));

static constexpr int Nn  = 512;
static constexpr int NNp = Nn * Nn;      // 262144
static constexpr int Cc  = 16;
static constexpr int Kk  = 262144;
static constexpr int KC  = 128;          // K-chunk staged in LDS per panel

// ---------------------------------------------------------------------------
// Kernel 0: zero grid planes and output
// ---------------------------------------------------------------------------
__global__ void nufft_zero_kernel(float* __restrict__ gr, float* __restrict__ gi,
                                  float* __restrict__ out) {
    int idx = blockIdx.x * blockDim.x + threadIdx.x;   // C*NN threads
    if (idx < Cc * NNp) {
        gr[idx] = 0.0f;
        gi[idx] = 0.0f;
    }
    if (idx < 2 * NNp) out[idx] = 0.0f;
}

// ---------------------------------------------------------------------------
// Kernel 1: DFT twiddle matrix W[p][u] = exp(2*pi*i*(p-256)*(u-256)/512)
// (ifftshift/fftshift folded in; W is symmetric)
// ---------------------------------------------------------------------------
__global__ void nufft_twiddle_kernel(float* __restrict__ Wr, float* __restrict__ Wi) {
    int idx = blockIdx.x * blockDim.x + threadIdx.x;   // NN threads
    if (idx >= NNp) return;
    int p = idx >> 9;
    int u = idx & 511;
    int t = ((p - 256) * (u - 256)) & 511;             // exact integer phase mod 512
    float ang = (float)t * 0.01227184630308513f;       // 2*pi/512
    float s, c;
    __sincosf(ang, &s, &c);
    Wr[idx] = c;
    Wi[idx] = s;
}

// ---------------------------------------------------------------------------
// Kernel 2: gridding (scatter with f32 global atomics)
// ---------------------------------------------------------------------------
__global__ void nufft_grid_kernel(const float* __restrict__ xr, const float* __restrict__ xi,
                                  const float* __restrict__ traj, const float* __restrict__ dcf,
                                  float* __restrict__ gr, float* __restrict__ gi) {
    int k = blockIdx.x * blockDim.x + threadIdx.x;
    if (k >= Kk) return;
    float ty = traj[2 * k + 0];
    float tx = traj[2 * k + 1];
    int iy = ((int)rintf((ty + 0.5f) * (float)Nn)) & (Nn - 1);  // rint == round-half-even
    int ix = ((int)rintf((tx + 0.5f) * (float)Nn)) & (Nn - 1);
    float w = dcf[k];
    int o = iy * Nn + ix;
#pragma unroll
    for (int c = 0; c < Cc; ++c) {
        atomicAdd(&gr[c * NNp + o], xr[c * Kk + k] * w);
        atomicAdd(&gi[c * NNp + o], xi[c * Kk + k] * w);
    }
}

// ---------------------------------------------------------------------------
// TDM: stage a KCx16 f32 panel (row stride Nn) from global into LDS.
// D# packing per CDNA5 ISA sec 8.3/8.4: group0 {count=1, lds_addr,
// global_addr, type=2}; group1 {data_size=4B, tensor dims, tile 16xKC,
// dim0 stride}. Groups 2/3 zero (<=2D tensor).
// ---------------------------------------------------------------------------
#if __has_builtin(__builtin_amdgcn_tensor_load_to_lds)
#define HAVE_TDM 1
#else
#define HAVE_TDM 0
#endif

__device__ __forceinline__ void tdm_panel_load(const float* gsrc, float* lds_dst,
                                               int rows) {
    unsigned long long ga = (unsigned long long)(uintptr_t)gsrc;
    unsigned int lds_off = (unsigned int)(uintptr_t)lds_dst;  // low 32 bits = LDS byte offset

    u32x4 g0;
    g0.x = 1u;                                            // count=1, user descriptor
    g0.y = lds_off;                                       // lds_addr
    g0.z = (unsigned int)ga;                              // global_addr[31:0]
    g0.w = (unsigned int)((ga >> 32) & 0x1FFFFFFull) | (2u << 30);  // addr[56:32] | type=2

    i32x8 g1;
    g1[0] = (2 << 16);                                    // data_size = 4 bytes
    g1[1] = (int)((unsigned)(Nn & 0xFFFF) << 16);         // tensor_dim0[15:0]
    g1[2] = (int)((unsigned)(Nn & 0xFFFF) << 16);         // dim0[31:16]=0 | tensor_dim1[15:0]
    g1[3] = (int)(16u << 16);                             // dim1[31:16]=0 | tile_dim0=16
    g1[4] = rows;                                         // tile_dim1 = rows, tile_dim2=0
    g1[5] = Nn;                                           // tensor_dim0_stride[31:0]
    g1[6] = 0;
    g1[7] = 0;

    i32x4 gz = {0, 0, 0, 0};
#if HAVE_TDM
#if __clang_major__ >= 23
    i32x8 gz8 = {0, 0, 0, 0, 0, 0, 0, 0};
    __builtin_amdgcn_tensor_load_to_lds(g0, g1, gz, gz, gz8, 0);
#else
    __builtin_amdgcn_tensor_load_to_lds(g0, g1, gz, gz, 0);
#endif
#else
    (void)g0; (void)g1; (void)gz;
#endif
}

__device__ __forceinline__ void tdm_wait0() {
#if __has_builtin(__builtin_amdgcn_s_wait_tensorcnt)
    __builtin_amdgcn_s_wait_tensorcnt(0);
#else
    asm volatile("s_wait_tensorcnt 0x0" ::: "memory");
#endif
}

// ---------------------------------------------------------------------------
// WMMA fragment helpers (V_WMMA_F32_16X16X4_F32 layouts, ISA 7.12.2)
// ---------------------------------------------------------------------------
__device__ __forceinline__ v2f load_frag_a(const float* __restrict__ A, int m0, int k0,
                                           int half, int l15) {
    const float* p = A + (m0 + l15) * Nn + k0 + 2 * half;
    v2f r; r.x = p[0]; r.y = p[1];
    return r;
}
// B fragment out of the LDS panel (row-major KCx16)
__device__ __forceinline__ v2f load_frag_b_lds(const float* lp, int k0, int half, int l15) {
    const float* p = lp + (k0 + 2 * half) * 16 + l15;
    v2f r; r.x = p[0]; r.y = p[16];
    return r;
}

#define WMMA_F32(a, b, c) \
    __builtin_amdgcn_wmma_f32_16x16x4_f32(false, (a), false, (b), (short)0, (c), false, false)

// ---------------------------------------------------------------------------
// Shared inner loop: one block = 8 waves covering a 128(M)x16(N) strip.
// B panel (KCx16, real+imag) staged in LDS via TDM (wave 0), fragments read
// with ds loads; A fragments are contiguous global b64 (L2 resident).
// ---------------------------------------------------------------------------
struct TileAcc { v8f rr, ri, ir, ii; };

__device__ __forceinline__ void complex_panel_mm(const float* __restrict__ Ar,
                                                 const float* __restrict__ Ai,
                                                 const float* __restrict__ Br,
                                                 const float* __restrict__ Bi,
                                                 float* lbR, float* lbI,
                                                 int m0, int n0, int wave,
                                                 int half, int l15, TileAcc& acc) {
    for (int kc = 0; kc < Nn; kc += KC) {
        if (wave == 0) {
            tdm_panel_load(Br + kc * Nn + n0, lbR, KC);
            tdm_panel_load(Bi + kc * Nn + n0, lbI, KC);
            tdm_wait0();
        }
#if !HAVE_TDM
        // fallback: cooperative vectorized staging
        for (int j = threadIdx.x; j < 2 * KC * 4; j += blockDim.x) {
            int plane = j >= KC * 4;
            int idx = plane ? j - KC * 4 : j;
            int r = idx >> 2, q = (idx & 3) << 2;
            const float* s = (plane ? Bi : Br) + (kc + r) * Nn + n0 + q;
            float* d = (plane ? lbI : lbR) + r * 16 + q;
            d[0] = s[0]; d[1] = s[1]; d[2] = s[2]; d[3] = s[3];
        }
#endif
        __syncthreads();
#pragma unroll 4
        for (int k0 = 0; k0 < KC; k0 += 4) {
            v2f ar = load_frag_a(Ar, m0, kc + k0, half, l15);
            v2f ai = load_frag_a(Ai, m0, kc + k0, half, l15);
            v2f br = load_frag_b_lds(lbR, k0, half, l15);
            v2f bi = load_frag_b_lds(lbI, k0, half, l15);
            acc.rr = WMMA_F32(ar, br, acc.rr);
            acc.ii = WMMA_F32(ai, bi, acc.ii);
            acc.ri = WMMA_F32(ar, bi, acc.ri);
            acc.ir = WMMA_F32(ai, br, acc.ir);
        }
        __syncthreads();
    }
}

// ---------------------------------------------------------------------------
// Kernel 3: pass A — T = W * g  (complex 512x512 matmul per coil)
// ---------------------------------------------------------------------------
__global__ void __launch_bounds__(256)
nufft_dft_rows_kernel(const float* __restrict__ Wr, const float* __restrict__ Wi,
                      const float* __restrict__ gr, const float* __restrict__ gi,
                      float* __restrict__ Tr, float* __restrict__ Ti) {
    __shared__ float lbR[KC * 16];
    __shared__ float lbI[KC * 16];
    int c    = blockIdx.y;
    int wave = threadIdx.x >> 5;
    int lane = threadIdx.x & 31;
    int mb   = blockIdx.x >> 5;             // 4 M-blocks of 128 rows
    int ns   = blockIdx.x & 31;             // 32 N-strips of 16 cols
    int m0   = mb * 128 + wave * 16;
    int n0   = ns * 16;
    int half = lane >> 4;
    int l15  = lane & 15;

    TileAcc acc; acc.rr = {}; acc.ri = {}; acc.ir = {}; acc.ii = {};
    complex_panel_mm(Wr, Wi, gr + c * NNp, gi + c * NNp, lbR, lbI,
                     m0, n0, wave, half, l15, acc);

    float* tR = Tr + c * NNp;
    float* tI = Ti + c * NNp;
    int col = n0 + l15;
#pragma unroll
    for (int r = 0; r < 8; ++r) {
        int row = m0 + r + 8 * half;
        tR[row * Nn + col] = acc.rr[r] - acc.ii[r];
        tI[row * Nn + col] = acc.ri[r] + acc.ir[r];
    }
}

// ---------------------------------------------------------------------------
// Kernel 4: pass B — img = T * W (W symmetric); fused conj(csm)*img reduction
// ---------------------------------------------------------------------------
__global__ void __launch_bounds__(256)
nufft_dft_cols_kernel(const float* __restrict__ Wr, const float* __restrict__ Wi,
                      const float* __restrict__ Tr, const float* __restrict__ Ti,
                      const float* __restrict__ csr, const float* __restrict__ csi,
                      float* __restrict__ out) {
    __shared__ float lbR[KC * 16];
    __shared__ float lbI[KC * 16];
    int c    = blockIdx.y;
    int wave = threadIdx.x >> 5;
    int lane = threadIdx.x & 31;
    int mb   = blockIdx.x >> 5;
    int ns   = blockIdx.x & 31;
    int m0   = mb * 128 + wave * 16;
    int n0   = ns * 16;
    int half = lane >> 4;
    int l15  = lane & 15;

    TileAcc acc; acc.rr = {}; acc.ri = {}; acc.ir = {}; acc.ii = {};
    complex_panel_mm(Tr + c * NNp, Ti + c * NNp, Wr, Wi, lbR, lbI,
                     m0, n0, wave, half, l15, acc);

    const float* cR = csr + c * NNp;
    const float* cI = csi + c * NNp;
    int col = n0 + l15;
#pragma unroll
    for (int r = 0; r < 8; ++r) {
        int row = m0 + r + 8 * half;
        float ir = acc.rr[r] - acc.ii[r];    // img real
        float ii = acc.ri[r] + acc.ir[r];    // img imag
        float a  = cR[row * Nn + col];
        float b  = cI[row * Nn + col];
        // conj(csm)*img = (a - i b)(ir + i ii)
        atomicAdd(&out[row * Nn + col],        a * ir + b * ii);
        atomicAdd(&out[NNp + row * Nn + col],  a * ii - b * ir);
    }
}

// ---------------------------------------------------------------------------
// Launcher
// ---------------------------------------------------------------------------
extern "C" void kernel_launch(void* const* d_in, const int* in_sizes, int n_in,
                              void* d_out, int out_size, void* d_ws, size_t ws_size,
                              hipStream_t stream) {
    const float* x_real   = (const float*)d_in[0];
    const float* x_imag   = (const float*)d_in[1];
    const float* csm_real = (const float*)d_in[2];
    const float* csm_imag = (const float*)d_in[3];
    const float* traj     = (const float*)d_in[4];
    const float* dcf      = (const float*)d_in[5];
    float* out = (float*)d_out;

    float* ws = (float*)d_ws;
    float* Wr = ws;                    // NN
    float* Wi = Wr + NNp;              // NN
    float* gr = Wi + NNp;              // C*NN
    float* gi = gr + Cc * NNp;         // C*NN
    float* Tr = gi + Cc * NNp;         // C*NN
    float* Ti = Tr + Cc * NNp;         // C*NN  (total ~69 MB)

    nufft_zero_kernel<<<(Cc * NNp + 255) / 256, 256, 0, stream>>>(gr, gi, out);
    nufft_twiddle_kernel<<<NNp / 256, 256, 0, stream>>>(Wr, Wi);
    nufft_grid_kernel<<<Kk / 256, 256, 0, stream>>>(x_real, x_imag, traj, dcf, gr, gi);

    dim3 gA(128, Cc);   // (4 M-blocks x 32 N-strips) x 16 coils
    nufft_dft_rows_kernel<<<gA, 256, 0, stream>>>(Wr, Wi, gr, gi, Tr, Ti);
    nufft_dft_cols_kernel<<<gA, 256, 0, stream>>>(Wr, Wi, Tr, Ti, csm_real, csm_imag, out);
}